// RGCN_4088808866420
// MI455X (gfx1250) — compile-verified
//
#include <hip/hip_runtime.h>
#include <hip/hip_bf16.h>

typedef __attribute__((ext_vector_type(2))) float v2f;
typedef __attribute__((ext_vector_type(8))) float v8f;

// ---------------------------------------------------------------------------
// Utility kernels
// ---------------------------------------------------------------------------
__global__ void fill_zero_kernel(float* __restrict__ p, size_t n) {
    size_t t = (size_t)blockIdx.x * blockDim.x + threadIdx.x;
    if (t < n) p[t] = 0.0f;
}

// Count degrees: deg_out[r*N + src], deg_in[r*N + dst] over 3*E edges.
__global__ void count_deg_kernel(const int* __restrict__ src, const int* __restrict__ dst,
                                 float* __restrict__ deg_out, float* __restrict__ deg_in,
                                 int N, int E) {
    int t = blockIdx.x * blockDim.x + threadIdx.x;
    if (t >= 3 * E) return;
    int r = t / E;
    unsafeAtomicAdd(&deg_out[(size_t)r * N + src[t]], 1.0f);
    unsafeAtomicAdd(&deg_in [(size_t)r * N + dst[t]], 1.0f);
}

__global__ void rsqrt_deg_kernel(float* __restrict__ p, size_t n) {
    size_t t = (size_t)blockIdx.x * blockDim.x + threadIdx.x;
    if (t < n) p[t] = rsqrtf(fmaxf(p[t], 1.0f));
}

// out[i*DO + c] = b[0][c] + b[1][c] + b[2][c]   (accumulator init = summed bias)
__global__ void init_bias_kernel(float* __restrict__ out, const float* __restrict__ b,
                                 int N, int DO) {
    size_t t = (size_t)blockIdx.x * blockDim.x + threadIdx.x;
    if (t >= (size_t)N * DO) return;
    int c = (int)(t % DO);
    out[t] = b[c] + b[DO + c] + b[2 * DO + c];
}

// ---------------------------------------------------------------------------
// WMMA GEMM:  m[N x DO] = (opt_relu(h[N x 128]) * invout[row]) @ W[128 x DO]
// Wave owns a fixed 16-col tile; B fragments (K=128) live in registers;
// streams 8 M-tiles of 16 rows.  fp32 exact via V_WMMA_F32_16X16X4_F32.
// ---------------------------------------------------------------------------
template <int DO, bool RELU>
__global__ __launch_bounds__(256) void rgcn_gemm_kernel(
    const float* __restrict__ h, const float* __restrict__ W,
    const float* __restrict__ invout, float* __restrict__ m, int n_mtiles) {
    constexpr int DI = 128;
    constexpr int NT = DO / 16;   // column tiles per row-band: 8 or 4
    constexpr int MG = 8 / NT;    // m-groups per block: 1 or 2
    const int wave = threadIdx.x >> 5;
    const int lane = threadIdx.x & 31;
    const int l16  = lane & 15;
    const int sel  = (lane >> 4) << 1;       // 0 (lanes 0-15) or 2 (lanes 16-31)
    const int ncol = (wave % NT) * 16;       // this wave's output column base
    const int mg   = wave / NT;

    // --- B fragments: W[k][ncol + l16] for all K, per ISA f32 B layout ---
    v2f bf[DI / 4];
#pragma unroll
    for (int kk = 0; kk < DI / 4; ++kk) {
        int k = kk * 4 + sel;
        bf[kk].x = W[(size_t)k * DO + ncol + l16];
        bf[kk].y = W[(size_t)(k + 1) * DO + ncol + l16];
    }

    const int mt0 = (blockIdx.x * MG + mg) * 8;
#pragma unroll 1
    for (int t = 0; t < 8; ++t) {
        int mtile = mt0 + t;
        if (mtile >= n_mtiles) break;        // uniform per wave: EXEC stays full
        int row = mtile * 16 + l16;
        float s = invout[row];
        const float* hrow = h + (size_t)row * DI;
        v8f acc = {};
#pragma unroll
        for (int kk = 0; kk < DI / 4; ++kk) {
            int k = kk * 4 + sel;
            v2f a;
            a.x = hrow[k];
            a.y = hrow[k + 1];
            if (RELU) { a.x = fmaxf(a.x, 0.0f); a.y = fmaxf(a.y, 0.0f); }
            a.x *= s; a.y *= s;
            acc = __builtin_amdgcn_wmma_f32_16x16x4_f32(
                false, a, false, bf[kk], (short)0, acc, false, false);
        }
        // C/D layout: VGPR i -> row = 16*mtile + i + (lane>=16 ? 8 : 0), col = ncol + l16
        int rbase = mtile * 16 + ((lane >> 4) << 3);
#pragma unroll
        for (int i = 0; i < 8; ++i)
            m[(size_t)(rbase + i) * DO + ncol + l16] = acc[i];
    }
}

// Scalar remainder GEMM for rows not covered by full 16-row tiles (N%16).
__global__ void rgcn_gemm_rem_kernel(const float* __restrict__ h, const float* __restrict__ W,
                                     const float* __restrict__ invout, float* __restrict__ m,
                                     int row0, int nrows, int DO, int relu) {
    int t = blockIdx.x * blockDim.x + threadIdx.x;
    if (t >= nrows * DO) return;
    int r = row0 + t / DO, c = t % DO;
    float s = invout[r];
    float acc = 0.0f;
    for (int k = 0; k < 128; ++k) {
        float hv = h[(size_t)r * 128 + k];
        if (relu) hv = fmaxf(hv, 0.0f);
        acc += hv * s * W[(size_t)k * DO + c];
    }
    m[(size_t)r * DO + c] = acc;
}

// ---------------------------------------------------------------------------
// Scatter-add:  out[dst] += m[src] * inv_in[dst]   (one wave per edge)
// ---------------------------------------------------------------------------
template <int DO>
__global__ __launch_bounds__(256) void rgcn_scatter_kernel(
    const float* __restrict__ m, const int* __restrict__ src, const int* __restrict__ dst,
    const float* __restrict__ invin, float* __restrict__ out, int E) {
    constexpr int VPT = DO / 32;  // 4 or 2 floats per lane
    int wid  = (blockIdx.x * blockDim.x + threadIdx.x) >> 5;
    int lane = threadIdx.x & 31;
    if (wid >= E) return;         // uniform per wave
    int s = src[wid], d = dst[wid];
    float sc = invin[d];
    const float* mp = m + (size_t)s * DO + lane * VPT;
    float* op = out + (size_t)d * DO + lane * VPT;
    if (VPT == 4) {
        float4 v = *(const float4*)mp;
        unsafeAtomicAdd(op + 0, v.x * sc);
        unsafeAtomicAdd(op + 1, v.y * sc);
        unsafeAtomicAdd(op + 2, v.z * sc);
        unsafeAtomicAdd(op + 3, v.w * sc);
    } else {
        float2 v = *(const float2*)mp;
        unsafeAtomicAdd(op + 0, v.x * sc);
        unsafeAtomicAdd(op + 1, v.y * sc);
    }
}

// ---------------------------------------------------------------------------
// Host orchestration
// ---------------------------------------------------------------------------
static inline int ceil_div(long long a, long long b) { return (int)((a + b - 1) / b); }

extern "C" void kernel_launch(void* const* d_in, const int* in_sizes, int n_in,
                              void* d_out, int out_size, void* d_ws, size_t ws_size,
                              hipStream_t stream) {
    const float* x   = (const float*)d_in[0];
    const int*   src = (const int*)d_in[1];
    const int*   dst = (const int*)d_in[2];
    const int N = in_sizes[0] / 128;
    const int E = in_sizes[1] / 3;

    float* ws      = (float*)d_ws;
    float* inv_out = ws;                                   // [3][N]
    float* inv_in  = ws + 3 * (size_t)N;                   // [3][N]
    float* hA      = ws + 6 * (size_t)N;                   // [N][128]
    float* hB      = hA + (size_t)N * 128;                 // [N][128]
    float* mbuf    = hB + (size_t)N * 128;                 // [N][128]

    // --- degree norms (computed once) ---
    size_t degn = 6 * (size_t)N;
    fill_zero_kernel<<<ceil_div(degn, 256), 256, 0, stream>>>(inv_out, degn);
    count_deg_kernel<<<ceil_div(3LL * E, 256), 256, 0, stream>>>(src, dst, inv_out, inv_in, N, E);
    rsqrt_deg_kernel<<<ceil_div(degn, 256), 256, 0, stream>>>(inv_out, degn);

    const int full = N / 16, rem = N - full * 16;
    const float* h_in = x;

    for (int L = 0; L < 6; ++L) {
        const float* W = (const float*)d_in[3 + 2 * L];
        const float* b = (const float*)d_in[4 + 2 * L];
        const int DO = (L == 5) ? 64 : 128;
        float* out = (L == 5) ? (float*)d_out : ((L & 1) ? hB : hA);

        init_bias_kernel<<<ceil_div((long long)N * DO, 256), 256, 0, stream>>>(out, b, N, DO);

        for (int r = 0; r < 3; ++r) {
            const float* Wr   = W + (size_t)r * 128 * DO;
            const float* ior  = inv_out + (size_t)r * N;
            const float* iir  = inv_in  + (size_t)r * N;
            // GEMM: m = (relu?(h) * inv_out) @ Wr
            if (L == 5) {
                rgcn_gemm_kernel<64, true><<<ceil_div(full, 16), 256, 0, stream>>>(
                    h_in, Wr, ior, mbuf, full);
            } else if (L == 0) {
                rgcn_gemm_kernel<128, false><<<ceil_div(full, 8), 256, 0, stream>>>(
                    h_in, Wr, ior, mbuf, full);
            } else {
                rgcn_gemm_kernel<128, true><<<ceil_div(full, 8), 256, 0, stream>>>(
                    h_in, Wr, ior, mbuf, full);
            }
            if (rem > 0) {
                rgcn_gemm_rem_kernel<<<ceil_div((long long)rem * DO, 256), 256, 0, stream>>>(
                    h_in, Wr, ior, mbuf, full * 16, rem, DO, L > 0 ? 1 : 0);
            }
            // Scatter: out[dst] += m[src] * inv_in[dst]
            if (DO == 128) {
                rgcn_scatter_kernel<128><<<ceil_div(E, 8), 256, 0, stream>>>(
                    mbuf, src + (size_t)r * E, dst + (size_t)r * E, iir, out, E);
            } else {
                rgcn_scatter_kernel<64><<<ceil_div(E, 8), 256, 0, stream>>>(
                    mbuf, src + (size_t)r * E, dst + (size_t)r * E, iir, out, E);
            }
        }
        h_in = out;
    }
}